// MultiScalePatchConvolution_84104049590447
// MI455X (gfx1250) — compile-verified
//
#include <hip/hip_runtime.h>
#include <math.h>

// Problem constants (from reference): B=256, S=16384, PATCH=16 -> 262144 patches, 64 channels.
#define NPATCHES (256 * 1024)
#define NCH 64
#define EPSV 1e-5f

typedef float v2f __attribute__((ext_vector_type(2)));
typedef float v8f __attribute__((ext_vector_type(8)));

// ---------------------------------------------------------------------------
// Kernel 0: zero the 9 (padded to 16) global moment accumulators in workspace.
// ---------------------------------------------------------------------------
__global__ void msp_zero(float* __restrict__ mom) {
    if (threadIdx.x < 16) mom[threadIdx.x] = 0.0f;
}

// ---------------------------------------------------------------------------
// Kernel 1: per-patch scalars (S, x_first, x_last, 1) -> pvec[patch] (float4),
// plus global reduction of the 9 first/second moments needed for BatchNorm.
// 1 thread = 1 patch. 1024 blocks x 256 threads.
// ---------------------------------------------------------------------------
__global__ void msp_pass1(const float* __restrict__ x,
                          float4* __restrict__ pvec,
                          float* __restrict__ mom) {
    const int patch = blockIdx.x * blockDim.x + threadIdx.x;
    const float4* x4 = (const float4*)x;
    const float4 v0 = x4[patch * 4 + 0];
    const float4 v1 = x4[patch * 4 + 1];
    const float4 v2 = x4[patch * 4 + 2];
    const float4 v3 = x4[patch * 4 + 3];
    const float S = (v0.x + v0.y + v0.z + v0.w) + (v1.x + v1.y + v1.z + v1.w) +
                    (v2.x + v2.y + v2.z + v2.w) + (v3.x + v3.y + v3.z + v3.w);
    const float xf = v0.x;
    const float xl = v3.w;
    pvec[patch] = make_float4(S, xf, xl, 1.0f);

    float p[9] = {S, xf, xl, S * S, xf * xf, xl * xl, S * xf, S * xl, xf * xl};

    __shared__ float acc[9];
    if (threadIdx.x < 9) acc[threadIdx.x] = 0.0f;
    __syncthreads();
#pragma unroll
    for (int i = 0; i < 9; ++i) {
        float v = p[i];
#pragma unroll
        for (int off = 16; off > 0; off >>= 1) v += __shfl_down(v, off, 32);
        if ((threadIdx.x & 31) == 0) atomicAdd(&acc[i], v);
    }
    __syncthreads();
    if (threadIdx.x < 9) atomicAdd(&mom[threadIdx.x], acc[threadIdx.x]);
}

// ---------------------------------------------------------------------------
// Kernel 2: fold conv-mean + BatchNorm(training stats) + gamma/beta into a
// single 4x64 coefficient matrix: out = a'*S + b'*xf + c'*xl + d'.
// Row k of coef multiplies A-feature k (K=0:S, K=1:xf, K=2:xl, K=3:1).
// COLUMN PERMUTATION: channel c lives at B-column (c%4)*16 + c/4, i.e.
// tile t, column n -> channel 4n + t. After the 4 WMMAs in pass 3 each lane
// then holds 4 consecutive channels per row -> one b128 store per row.
// One block, 64 threads (1 thread = 1 channel).
// ---------------------------------------------------------------------------
__global__ void msp_pass2(const float* __restrict__ mom,
                          const float* __restrict__ w2, const float* __restrict__ b2,
                          const float* __restrict__ w3, const float* __restrict__ b3,
                          const float* __restrict__ gamma, const float* __restrict__ beta,
                          float* __restrict__ coef) {
    const int c = threadIdx.x;
    if (c >= NCH) return;
    const float invN = 1.0f / (float)NPATCHES;
    const float mS = mom[0] * invN, mf = mom[1] * invN, ml = mom[2] * invN;
    const float cSS = mom[3] * invN - mS * mS;
    const float cff = mom[4] * invN - mf * mf;
    const float cll = mom[5] * invN - ml * ml;
    const float cSf = mom[6] * invN - mS * mf;
    const float cSl = mom[7] * invN - mS * ml;
    const float cfl = mom[8] * invN - mf * ml;

    float a, bc, cc, d;
    if (c < 32) {
        // ks=2, pad=1, L_out=17: both taps cover full patch sum.
        a  = (w2[c * 2 + 0] + w2[c * 2 + 1]) * (1.0f / 17.0f);
        bc = 0.0f;
        cc = 0.0f;
        d  = b2[c];
    } else {
        // ks=3, pad=1, L_out=16 (cross-correlation, XLA/torch orientation):
        // tap0 covers S - x_last, tap1 covers S, tap2 covers S - x_first.
        const int q = c - 32;
        const float t0 = w3[q * 3 + 0], t1 = w3[q * 3 + 1], t2 = w3[q * 3 + 2];
        a  = (t0 + t1 + t2) * (1.0f / 16.0f);
        bc = -t2 * (1.0f / 16.0f);   // coefficient on x_first
        cc = -t0 * (1.0f / 16.0f);   // coefficient on x_last
        d  = b3[q];
    }
    const float mean = a * mS + bc * mf + cc * ml + d;
    const float var  = a * a * cSS + bc * bc * cff + cc * cc * cll +
                       2.0f * (a * bc * cSf + a * cc * cSl + bc * cc * cfl);
    const float scale = gamma[c] * rsqrtf(var + EPSV);

    const int col = (c & 3) * 16 + (c >> 2);   // tile t = c%4, lane n = c/4
    coef[0 * NCH + col] = a * scale;
    coef[1 * NCH + col] = bc * scale;
    coef[2 * NCH + col] = cc * scale;
    coef[3 * NCH + col] = (d - mean) * scale + beta[c];
}

// ---------------------------------------------------------------------------
// Branchless exact-form GELU: 0.5*v*(1+erf(v/sqrt(2))) with erf computed via
// Abramowitz-Stegun 7.1.26 (max abs err 1.5e-7, at float-erff noise level):
//   erf(z) = 1 - (a1 t + a2 t^2 + a3 t^3 + a4 t^4 + a5 t^5) e^{-z^2},
//   t = 1/(1 + 0.3275911 z),  z >= 0;  odd extension via copysign.
// Straight-line: ~17 VALU ops, v_rcp_f32 + v_exp_f32, no EXEC divergence.
// ---------------------------------------------------------------------------
__device__ __forceinline__ float gelu_exact(float v) {
    const float z = fabsf(v) * 0.70710678118654752f;       // |v|/sqrt(2)
    const float t = __builtin_amdgcn_rcpf(fmaf(0.3275911f, z, 1.0f));
    float p = fmaf(1.061405429f, t, -1.453152027f);
    p = fmaf(p, t, 1.421413741f);
    p = fmaf(p, t, -0.284496736f);
    p = fmaf(p, t, 0.254829592f);
    p *= t;
    // e^{-z^2} = exp2(-z^2 * log2(e))
    const float e = p * __builtin_amdgcn_exp2f(-1.4426950408889634f * z * z);
    const float erfz = 1.0f - e;                           // erf(z), z >= 0
    const float s = copysignf(erfz, v);                    // erf(v/sqrt(2))
    return 0.5f * v * (1.0f + s);
}

// ---------------------------------------------------------------------------
// Kernel 3: main pass. One wave32 = 16 patches. A = 16x4 [S,xf,xl,1] per the
// V_WMMA_F32_16X16X4_F32 layout (lanes 0-15 hold K=0,1; lanes 16-31 K=2,3).
// Four back-to-back WMMAs cover 64 channels (permuted B tiles of 4x16), then
// branchless GELU and ONE float4 (b128) store per row per lane: lane n owns
// channels 4n..4n+3, so a half-wave writes 256 contiguous bytes per row.
// 2048 blocks x 256 threads (8 waves/block).
// ---------------------------------------------------------------------------
__global__ void msp_pass3(const float4* __restrict__ pvec,
                          const float* __restrict__ coef,
                          float* __restrict__ out) {
    const int lane  = threadIdx.x & 31;
    const int group = blockIdx.x * (blockDim.x >> 5) + (threadIdx.x >> 5);
    const int half  = lane >> 4;   // 0: K=0,1   1: K=2,3
    const int n     = lane & 15;   // row M for A, column N for B/D

    __builtin_prefetch(coef, 0, 0);  // global_prefetch_b8 of the tiny coef table

    const float4 pv = pvec[group * 16 + n];
    v2f a;
    a.x = half ? pv.z : pv.x;   // K = 2*half
    a.y = half ? pv.w : pv.y;   // K = 2*half + 1
    const int rb0 = (2 * half) * NCH;     // B vgpr0 row: K = 2*half
    const int rb1 = rb0 + NCH;            // B vgpr1 row: K = 2*half + 1

    v8f dt[4];
#pragma unroll
    for (int t = 0; t < 4; ++t) {
        v2f b;
        b.x = coef[rb0 + t * 16 + n];
        b.y = coef[rb1 + t * 16 + n];
        v8f cacc = {};
        dt[t] = __builtin_amdgcn_wmma_f32_16x16x4_f32(
            /*neg_a=*/false, a, /*neg_b=*/false, b,
            /*c_mod=*/(short)0, cacc, /*reuse_a=*/false, /*reuse_b=*/false);
    }

#pragma unroll
    for (int r = 0; r < 8; ++r) {
        const int M = r + 8 * half;           // C/D layout: vgpr r, half h -> M = r + 8h
        float4 g;
        g.x = gelu_exact(dt[0][r]);           // channel 4n + 0
        g.y = gelu_exact(dt[1][r]);           // channel 4n + 1
        g.z = gelu_exact(dt[2][r]);           // channel 4n + 2
        g.w = gelu_exact(dt[3][r]);           // channel 4n + 3
        *(float4*)(out + (size_t)(group * 16 + M) * NCH + 4 * n) = g;
    }
}

// ---------------------------------------------------------------------------
extern "C" void kernel_launch(void* const* d_in, const int* in_sizes, int n_in,
                              void* d_out, int out_size, void* d_ws, size_t ws_size,
                              hipStream_t stream) {
    (void)in_sizes; (void)n_in; (void)out_size; (void)ws_size;
    const float* x     = (const float*)d_in[0];
    const float* w2    = (const float*)d_in[1];
    const float* b2    = (const float*)d_in[2];
    const float* w3    = (const float*)d_in[3];
    const float* b3    = (const float*)d_in[4];
    const float* gamma = (const float*)d_in[5];
    const float* beta  = (const float*)d_in[6];
    float* out = (float*)d_out;

    float*  wsf  = (float*)d_ws;
    float*  mom  = wsf;                    // 16 floats
    float*  coef = wsf + 16;               // 256 floats
    float4* pvec = (float4*)(wsf + 1024);  // 4 MB, 16B-aligned offset

    msp_zero <<<1, 32, 0, stream>>>(mom);
    msp_pass1<<<NPATCHES / 256, 256, 0, stream>>>(x, pvec, mom);
    msp_pass2<<<1, 64, 0, stream>>>(mom, w2, b2, w3, b3, gamma, beta, coef);
    msp_pass3<<<NPATCHES / 16 / 8, 256, 0, stream>>>(pvec, coef, out);
}